// DavidBeansV2_88493506167414
// MI455X (gfx1250) — compile-verified
//
#include <hip/hip_runtime.h>
#include <stdint.h>

typedef __attribute__((ext_vector_type(2))) float v2f;
typedef __attribute__((ext_vector_type(4))) float v4f;
typedef __attribute__((ext_vector_type(8))) float v8f;
typedef __attribute__((ext_vector_type(4))) unsigned int u32x4;
typedef __attribute__((ext_vector_type(4))) int i32x4;
typedef __attribute__((ext_vector_type(8))) int i32x8;

#define WMMA_F32(A, B, C) \
  __builtin_amdgcn_wmma_f32_16x16x4_f32(false, (A), false, (B), (short)0, (C), false, false)

#if defined(__has_builtin)
#if __has_builtin(__builtin_amdgcn_tensor_load_to_lds) && \
    __has_builtin(__builtin_amdgcn_s_wait_tensorcnt)
#define USE_TDM 1
#endif
#endif

#define KC 64          // K-chunk staged per LDS buffer
#define LSTR 68        // padded LDS row stride (dwords): conflict-free banks
// LDS per buffer: 64 rows * 68 dwords = 17408 B; double-buffered = 34816 B.

#ifdef USE_TDM
// TDM: load a 64-row x KC-float tile (row pitch = `pitch` floats, K-contiguous
// rows) from global `gsrc` into LDS at byte offset `lds_off`, inserting a
// 4-dword pad after every 64 dwords so the LDS row stride becomes LSTR.
__device__ __forceinline__ void tdm_load_panel(const float* gsrc,
                                               unsigned lds_off, int pitch)
{
  const unsigned long long ga = (unsigned long long)(uintptr_t)gsrc;
  u32x4 g0;
  g0[0] = 1u;                                   // count=1, user descriptor
  g0[1] = lds_off;                              // lds_addr (bytes)
  g0[2] = (unsigned)ga;                         // global_addr[31:0]
  g0[3] = (unsigned)(ga >> 32) | (2u << 30);    // global_addr[56:32] | type=2
  i32x8 g1;
  g1[0] = (2 << 16) | (1 << 20) | (5 << 22) | (3 << 25);
           // data_size=4B | pad_enable | pad_interval=64dw | pad_amount=4dw
  g1[1] = (KC << 16);                           // tensor_dim0 = KC (lo16)
  g1[2] = (64 << 16);                           // tensor_dim1 = 64 rows (lo16)
  g1[3] = (KC << 16);                           // tile_dim0 = KC
  g1[4] = 64;                                   // tile_dim1 = 64, tile_dim2 = 0
  g1[5] = pitch;                                // tensor_dim0_stride (lo32)
  g1[6] = 0;
  g1[7] = 0;
  const i32x4 z4 = {0, 0, 0, 0};
#if __clang_major__ >= 23
  const i32x8 z8 = {0, 0, 0, 0, 0, 0, 0, 0};
  __builtin_amdgcn_tensor_load_to_lds(g0, g1, z4, z4, z8, 0);
#else
  __builtin_amdgcn_tensor_load_to_lds(g0, g1, z4, z4, 0);
#endif
}
#endif

// ---------------- 512x512 transpose (32x32 LDS tiles) -----------------------
__global__ __launch_bounds__(256) void transpose512(
    const float* __restrict__ in, float* __restrict__ out)
{
  __shared__ float t[32][33];
  const int tx = threadIdx.x & 31, ty = threadIdx.x >> 5;
  const int bx = (blockIdx.x & 15) << 5, by = (blockIdx.x >> 4) << 5;
  #pragma unroll
  for (int i = 0; i < 4; ++i)
    t[ty + 8 * i][tx] = in[(size_t)(by + ty + 8 * i) * 512 + bx + tx];
  __syncthreads();
  #pragma unroll
  for (int i = 0; i < 4; ++i)
    out[(size_t)(bx + ty + 8 * i) * 512 + by + tx] = t[tx][ty + 8 * i];
}

// -----------------------------------------------------------------------------
// C[M,512] = A[M,512] @ W + bias, where Wt = W^T (512x512, row-major in K).
// One block (8 waves) = 128 rows x 64 cols. The 64-col x KC-k panel of Wt is
// staged into LDS (K-major, LSTR stride) double-buffered via the TDM.
// B fragments are single aligned ds_load_b64; A is software-pipelined.
// A row m lives at A + (m>>8)*batchPitch + rowBase + (m&255)*512.
// -----------------------------------------------------------------------------
__global__ __launch_bounds__(256) void gemm_bias_f32(
    const float* __restrict__ A, const float* __restrict__ Wt,
    const float* __restrict__ bias, float* __restrict__ C,
    int batchPitch, int rowBase)
{
  __shared__ float Wlds[2][64 * LSTR];

  const int wave = threadIdx.x >> 5;
  const int lane = threadIdx.x & 31;
  const int half = lane >> 4;
  const int lm   = lane & 15;

  const int rowBlock = blockIdx.x >> 3;          // 128 rows per block
  const int colBlk   = blockIdx.x & 7;
  const int nBase = colBlk << 6;
  const int mBase = rowBlock * 128 + wave * 16;

  const int m = mBase + lm;
  const float* Ap = A + (size_t)(m >> 8) * (size_t)batchPitch
                      + rowBase + (size_t)(m & 255) * 512 + 2 * half;
  const float* Alast = Ap + (512 - 4);
  v2f a = *(const v2f*)Ap;

  v8f acc[4] = {v8f{}, v8f{}, v8f{}, v8f{}};
  int buf = 0;

#ifdef USE_TDM
  if (wave == 0)
    tdm_load_panel(Wt + (size_t)nBase * 512,
                   (unsigned)(uintptr_t)&Wlds[0][0], 512);
#endif

  for (int kc = 0; kc < 512; kc += KC) {
#ifdef USE_TDM
    if (wave == 0) __builtin_amdgcn_s_wait_tensorcnt(0);
    __syncthreads();
    if (wave == 0 && kc + KC < 512)
      tdm_load_panel(Wt + (size_t)nBase * 512 + kc + KC,
                     (unsigned)(uintptr_t)&Wlds[buf ^ 1][0], 512);
#else
    {
      const float* src = Wt + (size_t)nBase * 512 + kc;
      #pragma unroll
      for (int i = 0; i < 4; ++i) {              // 64x64 floats = 1024 float4
        const int f = threadIdx.x + (i << 8);
        const int r = f >> 4, q = (f & 15) << 2;
        *(v4f*)&Wlds[buf][r * LSTR + q] = *(const v4f*)(src + (size_t)r * 512 + q);
      }
    }
    __syncthreads();
#endif

    const float* lb = &Wlds[buf][lm * LSTR + 2 * half];
    #pragma unroll 4
    for (int kr = 0; kr < KC; kr += 4) {
      const float* An = Ap + 4;
      if (An > Alast) An = Alast;                // clamp final prefetch
      const v2f an = *(const v2f*)An;
      #pragma unroll
      for (int c = 0; c < 4; ++c) {
        const v2f b = *(const v2f*)(lb + 16 * LSTR * c + kr);  // ds_load_b64
        acc[c] = WMMA_F32(a, b, acc[c]);
      }
      a = an; Ap = An;
    }
    __syncthreads();
    buf ^= 1;
  }

  #pragma unroll
  for (int j = 0; j < 8; ++j) {
    const int row = mBase + j + 8 * half;
    float* Crow = C + (size_t)row * 512 + nBase + lm;
    #pragma unroll
    for (int c = 0; c < 4; ++c)
      Crow[16 * c] = acc[c][j] + bias[nBase + lm + 16 * c];
  }
}

// -----------------------------------------------------------------------------
// S[b] = q_r[b] @ k_r[b]^T + pos_bias, diagonal -> -1e9. k_r rows are already
// K-contiguous, so the B panel uses the identical TDM staging path.
// Block = one batch's 128-row x 64-col tile: 64 batches * 2 * 4 = 512 blocks.
// -----------------------------------------------------------------------------
__global__ __launch_bounds__(256) void scores_f32(
    const float* __restrict__ Qr, const float* __restrict__ Kr,
    const float* __restrict__ PB, float* __restrict__ S)
{
  __shared__ float Wlds[2][64 * LSTR];

  const int wave = threadIdx.x >> 5;
  const int lane = threadIdx.x & 31;
  const int half = lane >> 4;
  const int lm   = lane & 15;

  const int b  = blockIdx.x >> 3;
  const int t  = blockIdx.x & 7;
  const int mBase = (t >> 2) * 128 + wave * 16;  // row within batch
  const int nBase = (t & 3) << 6;

  const float* Kb = Kr + (size_t)b * 256 * 512;
  const float* Ap = Qr + ((size_t)b * 256 + mBase + lm) * 512 + 2 * half;
  const float* Alast = Ap + (512 - 4);
  v2f a = *(const v2f*)Ap;

  v8f acc[4] = {v8f{}, v8f{}, v8f{}, v8f{}};
  int buf = 0;

#ifdef USE_TDM
  if (wave == 0)
    tdm_load_panel(Kb + (size_t)nBase * 512,
                   (unsigned)(uintptr_t)&Wlds[0][0], 512);
#endif

  for (int kc = 0; kc < 512; kc += KC) {
#ifdef USE_TDM
    if (wave == 0) __builtin_amdgcn_s_wait_tensorcnt(0);
    __syncthreads();
    if (wave == 0 && kc + KC < 512)
      tdm_load_panel(Kb + (size_t)nBase * 512 + kc + KC,
                     (unsigned)(uintptr_t)&Wlds[buf ^ 1][0], 512);
#else
    {
      const float* src = Kb + (size_t)nBase * 512 + kc;
      #pragma unroll
      for (int i = 0; i < 4; ++i) {
        const int f = threadIdx.x + (i << 8);
        const int r = f >> 4, q = (f & 15) << 2;
        *(v4f*)&Wlds[buf][r * LSTR + q] = *(const v4f*)(src + (size_t)r * 512 + q);
      }
    }
    __syncthreads();
#endif

    const float* lb = &Wlds[buf][lm * LSTR + 2 * half];
    #pragma unroll 4
    for (int kr = 0; kr < KC; kr += 4) {
      const float* An = Ap + 4;
      if (An > Alast) An = Alast;
      const v2f an = *(const v2f*)An;
      #pragma unroll
      for (int c = 0; c < 4; ++c) {
        const v2f bfr = *(const v2f*)(lb + 16 * LSTR * c + kr);
        acc[c] = WMMA_F32(a, bfr, acc[c]);
      }
      a = an; Ap = An;
    }
    __syncthreads();
    buf ^= 1;
  }

  #pragma unroll
  for (int j = 0; j < 8; ++j) {
    const int row = mBase + j + 8 * half;
    float* Srow = S + ((size_t)b * 256 + row) * 256;
    const float* PBrow = PB + (size_t)row * 256;
    #pragma unroll
    for (int c = 0; c < 4; ++c) {
      const int col = nBase + 16 * c + lm;
      float v = acc[c][j] + PBrow[col];
      if (row == col) v = -1.0e9f;
      Srow[col] = v;
    }
  }
}

// ---------------- row-wise L2 normalize (one wave per 512-float row) --------
__global__ __launch_bounds__(256) void l2norm_rows(float* __restrict__ X)
{
  const int row  = blockIdx.x * 8 + (threadIdx.x >> 5);
  const int lane = threadIdx.x & 31;
  float* r = X + (size_t)row * 512;
  float s = 0.f;
  float v[16];
  #pragma unroll
  for (int i = 0; i < 16; ++i) { v[i] = r[lane + 32 * i]; s += v[i] * v[i]; }
  #pragma unroll
  for (int off = 16; off >= 1; off >>= 1) s += __shfl_xor(s, off, 32);
  const float inv = 1.0f / sqrtf(s + 1e-12f);
  #pragma unroll
  for (int i = 0; i < 16; ++i) r[lane + 32 * i] = v[i] * inv;
}

// ---------------- top-8 per 256-wide row (wave per row) ---------------------
__global__ __launch_bounds__(256) void topk8(
    const float* __restrict__ S, int* __restrict__ routes)
{
  const int row  = blockIdx.x * 8 + (threadIdx.x >> 5);
  const int lane = threadIdx.x & 31;
  const float* s = S + (size_t)row * 256;
  float v[8];
  #pragma unroll
  for (int i = 0; i < 8; ++i) v[i] = s[lane + 32 * i];

  for (int sel = 0; sel < 8; ++sel) {
    float bv = v[0]; int bi = lane;
    #pragma unroll
    for (int i = 1; i < 8; ++i)
      if (v[i] > bv) { bv = v[i]; bi = lane + 32 * i; }
    #pragma unroll
    for (int off = 16; off >= 1; off >>= 1) {
      const float ov = __shfl_xor(bv, off, 32);
      const int   oi = __shfl_xor(bi, off, 32);
      if (ov > bv || (ov == bv && oi < bi)) { bv = ov; bi = oi; }
    }
    if (lane == 0) routes[(size_t)row * 8 + sel] = bi;
    #pragma unroll
    for (int i = 0; i < 8; ++i)
      if (bi == lane + 32 * i) v[i] = -3.0e38f;
  }
}

// ---------------- gathered 8-way attention + context ------------------------
__global__ __launch_bounds__(256) void attn_ctx(
    const float* __restrict__ Q, const float* __restrict__ Kh,
    const float* __restrict__ Vh, const int* __restrict__ routes,
    float* __restrict__ ctx)
{
  const int bp   = blockIdx.x;
  const int h    = threadIdx.x >> 5;
  const int lane = threadIdx.x & 31;
  const int col  = h * 64 + lane * 2;
  const int b    = bp >> 8;

  const float* q = Q + (size_t)bp * 512 + col;
  const float qx = q[0], qy = q[1];

  int r[8];
  const int* rt = routes + (size_t)bp * 8;
  #pragma unroll
  for (int k = 0; k < 8; ++k) r[k] = rt[k];

  float logit[8];
  #pragma unroll
  for (int k = 0; k < 8; ++k) {
    const float* kk = Kh + ((size_t)b * 256 + r[k]) * 512 + col;
    float p = qx * kk[0] + qy * kk[1];
    #pragma unroll
    for (int off = 16; off >= 1; off >>= 1) p += __shfl_xor(p, off, 32);
    logit[k] = p * 0.125f;
  }

  float mx = logit[0];
  #pragma unroll
  for (int k = 1; k < 8; ++k) mx = fmaxf(mx, logit[k]);
  float w[8], den = 0.f;
  #pragma unroll
  for (int k = 0; k < 8; ++k) { w[k] = expf(logit[k] - mx); den += w[k]; }
  const float inv = 1.0f / den;

  float cx = 0.f, cy = 0.f;
  #pragma unroll
  for (int k = 0; k < 8; ++k) {
    const float* vv = Vh + ((size_t)b * 256 + r[k]) * 512 + col;
    const float aw = w[k] * inv;
    cx += aw * vv[0]; cy += aw * vv[1];
  }
  float* o = ctx + (size_t)bp * 512 + col;
  o[0] = cx; o[1] = cy;
}

// -----------------------------------------------------------------------------
extern "C" void kernel_launch(void* const* d_in, const int* in_sizes, int n_in,
                              void* d_out, int out_size, void* d_ws, size_t ws_size,
                              hipStream_t stream)
{
  const float* x    = (const float*)d_in[0];
  const float* wq_r = (const float*)d_in[1];
  const float* bq_r = (const float*)d_in[2];
  const float* wk_r = (const float*)d_in[3];
  const float* bk_r = (const float*)d_in[4];
  const float* pb   = (const float*)d_in[5];
  const float* wq   = (const float*)d_in[6];
  const float* bq   = (const float*)d_in[7];
  const float* wk   = (const float*)d_in[8];
  const float* bk   = (const float*)d_in[9];
  const float* wv   = (const float*)d_in[10];
  const float* bv   = (const float*)d_in[11];
  const float* wo   = (const float*)d_in[12];
  const float* bo   = (const float*)d_in[13];

  const int B = 64, P = 256, D = 512;
  const size_t MD = (size_t)B * P * D;
  const size_t WD = (size_t)D * D;

  float* qr   = (float*)d_ws;
  float* kr   = qr  + MD;
  float* qh   = kr  + MD;
  float* kh   = qh  + MD;
  float* vh   = kh  + MD;
  float* ctx  = vh  + MD;
  float* sc   = ctx + MD;                      // B*P*P floats
  int*   rts  = (int*)(sc + (size_t)B * P * P);
  float* wT   = (float*)(rts + (size_t)B * P * 8);
  float* wqrT = wT;           float* wkrT = wT + WD;
  float* wqT  = wT + 2 * WD;  float* wkT  = wT + 3 * WD;
  float* wvT  = wT + 4 * WD;  float* woT  = wT + 5 * WD;

  const dim3 blk(256);
  const int xPitch = (P + 1) * D;                       // skip x[:,0,:]
  const int gemmBlocks = (B * P / 128) * (D / 64);      // 1024

  // 0) transpose weights once (K-contiguous panels for the TDM)
  transpose512<<<256, blk, 0, stream>>>(wq_r, wqrT);
  transpose512<<<256, blk, 0, stream>>>(wk_r, wkrT);
  transpose512<<<256, blk, 0, stream>>>(wq,   wqT);
  transpose512<<<256, blk, 0, stream>>>(wk,   wkT);
  transpose512<<<256, blk, 0, stream>>>(wv,   wvT);
  transpose512<<<256, blk, 0, stream>>>(wo,   woT);

  // 1) routing projections + L2 norm
  gemm_bias_f32<<<gemmBlocks, blk, 0, stream>>>(x, wqrT, bq_r, qr, xPitch, D);
  gemm_bias_f32<<<gemmBlocks, blk, 0, stream>>>(x, wkrT, bk_r, kr, xPitch, D);
  l2norm_rows<<<B * P / 8, blk, 0, stream>>>(qr);
  l2norm_rows<<<B * P / 8, blk, 0, stream>>>(kr);

  // 2) similarity + top-8 routes
  scores_f32<<<B * 8, blk, 0, stream>>>(qr, kr, pb, sc);
  topk8<<<B * P / 8, blk, 0, stream>>>(sc, rts);

  // 3) head projections
  gemm_bias_f32<<<gemmBlocks, blk, 0, stream>>>(x, wqT, bq, qh, xPitch, D);
  gemm_bias_f32<<<gemmBlocks, blk, 0, stream>>>(x, wkT, bk, kh, xPitch, D);
  gemm_bias_f32<<<gemmBlocks, blk, 0, stream>>>(x, wvT, bv, vh, xPitch, D);

  // 4) gathered attention -> ctx
  attn_ctx<<<B * P, blk, 0, stream>>>(qh, kh, vh, rts, ctx);

  // 5) output projection -> d_out
  gemm_bias_f32<<<gemmBlocks, blk, 0, stream>>>(ctx, woT, bo, (float*)d_out,
                                                P * D, 0);
}